// Filtering_46720654246501
// MI455X (gfx1250) — compile-verified
//
#include <hip/hip_runtime.h>

typedef __attribute__((ext_vector_type(2))) float v2f;
typedef __attribute__((ext_vector_type(8))) float v8f;

// Small per-block tables derived from the IIR coefficients (order 4, DF2T):
//   h[n]      : output impulse response taps 0..15
//   zrec[n]   : state after n steps of the x-impulse simulation (U columns)
//   S[m][k]   : y[m] response to unit initial state component k
//   M[j][k]   : state transition over 16 steps
struct BlkTab {
  float h[16];
  float zrec[17][4];
  float S[16][4];
  float M[4][4];
};

__device__ __forceinline__ v8f wmma4(v2f a, v2f b, v8f c) {
  // D(16x16,f32) = A(16x4,f32) x B(4x16,f32) + C
  return __builtin_amdgcn_wmma_f32_16x16x4_f32(false, a, false, b, (short)0, c,
                                               false, false);
}

__device__ void build_tables(const float* __restrict__ bc,
                             const float* __restrict__ ac, BlkTab& T) {
  if (threadIdx.x == 0) {
    float a0 = ac[0];
    float bb[5], aa[5];
#pragma unroll
    for (int i = 0; i < 5; ++i) { bb[i] = bc[i] / a0; aa[i] = ac[i] / a0; }
    // x-impulse simulation: gives h[] and zrec[] (-> T and U matrices)
    float z0 = 0.f, z1 = 0.f, z2 = 0.f, z3 = 0.f;
#pragma unroll
    for (int n = 0; n < 16; ++n) {
      float x = (n == 0) ? 1.0f : 0.0f;
      float y = bb[0] * x + z0;
      float n0 = bb[1] * x - aa[1] * y + z1;
      float n1 = bb[2] * x - aa[2] * y + z2;
      float n2 = bb[3] * x - aa[3] * y + z3;
      float n3 = bb[4] * x - aa[4] * y;
      z0 = n0; z1 = n1; z2 = n2; z3 = n3;
      T.h[n] = y;
      T.zrec[n + 1][0] = z0; T.zrec[n + 1][1] = z1;
      T.zrec[n + 1][2] = z2; T.zrec[n + 1][3] = z3;
    }
    // unit-initial-state simulations: gives S and M^16
    for (int k = 0; k < 4; ++k) {
      float w0 = (k == 0) ? 1.f : 0.f, w1 = (k == 1) ? 1.f : 0.f;
      float w2 = (k == 2) ? 1.f : 0.f, w3 = (k == 3) ? 1.f : 0.f;
#pragma unroll
      for (int n = 0; n < 16; ++n) {
        float y = w0;
        float n0 = -aa[1] * y + w1;
        float n1 = -aa[2] * y + w2;
        float n2 = -aa[3] * y + w3;
        float n3 = -aa[4] * y;
        w0 = n0; w1 = n1; w2 = n2; w3 = n3;
        T.S[n][k] = y;
      }
      T.M[0][k] = w0; T.M[1][k] = w1; T.M[2][k] = w2; T.M[3][k] = w3;
    }
  }
  __syncthreads();
}

// Gather per-lane A-operand fragments per CDNA5 16x4 f32 A layout:
// lane L (L<16): M=L, VGPR0=K0, VGPR1=K1 ; lane L (>=16): M=L-16, K2/K3.
__device__ __forceinline__ void gather_mats(const BlkTab& T, int m, int half,
                                            v2f Ta[4], v2f Ua[4], v2f& Sa,
                                            v2f& Ma) {
#pragma unroll
  for (int k = 0; k < 4; ++k) {
    int j0 = 4 * k + 2 * half;
    Ta[k][0] = (m >= j0)     ? T.h[m - j0]     : 0.0f;
    Ta[k][1] = (m >= j0 + 1) ? T.h[m - j0 - 1] : 0.0f;
    Ua[k][0] = (m < 4) ? T.zrec[16 - j0][m] : 0.0f; // U[m][j0]
    Ua[k][1] = (m < 4) ? T.zrec[15 - j0][m] : 0.0f; // U[m][j0+1]
  }
  int kc = 2 * half;
  Sa[0] = T.S[m][kc];
  Sa[1] = T.S[m][kc + 1];
  Ma[0] = (m < 4) ? T.M[m][kc] : 0.0f;
  Ma[1] = (m < 4) ? T.M[m][kc + 1] : 0.0f;
}

// ---------------- stats: mean / max|x-mean| per signal (one wave each) -----
__global__ void __launch_bounds__(256)
stats_kernel(const float* __restrict__ x, float* __restrict__ meanv,
             float* __restrict__ invv, float* __restrict__ mxv, int nsig,
             int nt) {
  int wave = blockIdx.x * (blockDim.x >> 5) + (threadIdx.x >> 5);
  if (wave >= nsig) return;
  int lane = threadIdx.x & 31;
  const float* xs = x + (size_t)wave * nt;
  float s = 0.0f, lo = 3.4e38f, hi = -3.4e38f;
  int n4 = nt >> 2;
  for (int i = lane; i < n4; i += 32) {
    float4 v = ((const float4*)xs)[i];
    s += (v.x + v.y) + (v.z + v.w);
    hi = fmaxf(hi, fmaxf(fmaxf(v.x, v.y), fmaxf(v.z, v.w)));
    lo = fminf(lo, fminf(fminf(v.x, v.y), fminf(v.z, v.w)));
  }
  for (int t = (nt & ~3) + lane; t < nt; t += 32) {
    float v = xs[t];
    s += v; hi = fmaxf(hi, v); lo = fminf(lo, v);
  }
#pragma unroll
  for (int off = 16; off >= 1; off >>= 1) {
    s += __shfl_xor(s, off, 32);
    hi = fmaxf(hi, __shfl_xor(hi, off, 32));
    lo = fminf(lo, __shfl_xor(lo, off, 32));
  }
  if (lane == 0) {
    float mu = s / (float)nt;
    float mx = fmaxf(hi - mu, mu - lo); // == max|x-mu| (monotone in x)
    if (mx == 0.0f) mx = 1.0f;
    meanv[wave] = mu;
    invv[wave] = 1.0f / mx;
    mxv[wave] = mx;
  }
}

// ---------------- forward filter: 16 signals x 16 timesteps per WMMA block --
__global__ void __launch_bounds__(128)
fwd_kernel(const float* __restrict__ x, const float* __restrict__ bc,
           const float* __restrict__ ac, const float* __restrict__ taper,
           const float* __restrict__ meanv, const float* __restrict__ invv,
           float* __restrict__ y, int nsig, int nt) {
  __shared__ BlkTab T;
  build_tables(bc, ac, T);

  int wave = blockIdx.x * (blockDim.x >> 5) + (threadIdx.x >> 5);
  if (wave * 16 >= nsig) return; // whole-wave guard; EXEC stays all-ones
  int lane = threadIdx.x & 31;
  int m = lane & 15;   // signal within tile (N), also A-matrix row (M)
  int half = lane >> 4;

  v2f Ta[4], Ua[4], Sa, Ma;
  gather_mats(T, m, half, Ta, Ua, Sa, Ma);

  int sig = wave * 16 + m;
  const float* xs = x + (size_t)sig * nt;
  float* ys = y + (size_t)sig * nt;
  float mu = meanv[sig];
  float iv = invv[sig];

  v2f Zb = {0.0f, 0.0f};
  for (int t0 = 0; t0 < nt; t0 += 16) {
    // Load X tile as four 4x16 B-matrix chunks (K rows = time, N = signal).
    v2f X[4];
#pragma unroll
    for (int k = 0; k < 4; ++k) {
      int tau = t0 + 4 * k + 2 * half;
      v2f xv = *(const v2f*)(xs + tau);
      v2f tp = *(const v2f*)(taper + tau);
      X[k][0] = (xv[0] - mu) * iv * tp[0];
      X[k][1] = (xv[1] - mu) * iv * tp[1];
    }
    // Z_out = U*X + M16*Z_in  (critical chain: only the last WMMA needs Zb)
    v8f zc = {};
#pragma unroll
    for (int k = 0; k < 4; ++k) zc = wmma4(Ua[k], X[k], zc);
    zc = wmma4(Ma, Zb, zc);
    // Y = T*X + S*Z_in
    v8f yc = {};
#pragma unroll
    for (int k = 0; k < 4; ++k) yc = wmma4(Ta[k], X[k], yc);
    yc = wmma4(Sa, Zb, yc);
    // Store Y (D layout: VGPR r -> time t0+r (lanes<16) / t0+8+r (lanes>=16))
    float* yp = ys + t0 + 8 * half;
#pragma unroll
    for (int r = 0; r < 8; ++r) yp[r] = yc[r];
    // Repack Z_out rows 0..3 (D layout) into B(4x16) layout for next block.
    float hz0 = __shfl(zc[2], m, 32);
    float hz1 = __shfl(zc[3], m, 32);
    Zb[0] = half ? hz0 : zc[0];
    Zb[1] = half ? hz1 : zc[1];
  }
}

// ---------------- backward filter: time-reversed, in place, x mx -----------
__global__ void __launch_bounds__(128)
bwd_kernel(const float* __restrict__ bc, const float* __restrict__ ac,
           const float* __restrict__ mxv, float* __restrict__ y, int nsig,
           int nt) {
  __shared__ BlkTab T;
  build_tables(bc, ac, T);

  int wave = blockIdx.x * (blockDim.x >> 5) + (threadIdx.x >> 5);
  if (wave * 16 >= nsig) return;
  int lane = threadIdx.x & 31;
  int m = lane & 15;
  int half = lane >> 4;

  v2f Ta[4], Ua[4], Sa, Ma;
  gather_mats(T, m, half, Ta, Ua, Sa, Ma);

  int sig = wave * 16 + m;
  float* ys = y + (size_t)sig * nt;
  float sc = mxv[sig];

  v2f Zb = {0.0f, 0.0f};
  for (int t0 = 0; t0 < nt; t0 += 16) {
    v2f X[4];
#pragma unroll
    for (int k = 0; k < 4; ++k) {
      int tau = t0 + 4 * k + 2 * half;
      // u[tau] = y[nt-1-tau]; load the pair (y[nt-2-tau], y[nt-1-tau]), swap.
      v2f r = *(const v2f*)(ys + (nt - 2 - tau));
      X[k][0] = r[1];
      X[k][1] = r[0];
    }
    v8f zc = {};
#pragma unroll
    for (int k = 0; k < 4; ++k) zc = wmma4(Ua[k], X[k], zc);
    zc = wmma4(Ma, Zb, zc);
    v8f yc = {};
#pragma unroll
    for (int k = 0; k < 4; ++k) yc = wmma4(Ta[k], X[k], yc);
    yc = wmma4(Sa, Zb, yc);
    // out[nt-1-t] = v[t]*mx ; in-place is safe: this wave read these
    // addresses earlier in this same iteration (program order preserved).
    float* op = ys + (nt - 1 - t0 - 8 * half);
#pragma unroll
    for (int r = 0; r < 8; ++r) op[-r] = yc[r] * sc;
    float hz0 = __shfl(zc[2], m, 32);
    float hz1 = __shfl(zc[3], m, 32);
    Zb[0] = half ? hz0 : zc[0];
    Zb[1] = half ? hz1 : zc[1];
  }
}

extern "C" void kernel_launch(void* const* d_in, const int* in_sizes, int n_in,
                              void* d_out, int out_size, void* d_ws,
                              size_t ws_size, hipStream_t stream) {
  const float* x = (const float*)d_in[0];
  const float* b = (const float*)d_in[1];
  const float* a = (const float*)d_in[2];
  const float* taper = (const float*)d_in[3];
  int nt = in_sizes[3];              // 6000 (multiple of 16)
  int nsig = in_sizes[0] / nt;       // 6144 (multiple of 16)

  float* ws = (float*)d_ws;
  float* meanv = ws;
  float* invv = ws + nsig;
  float* mxv = ws + 2 * nsig;
  float* y = (float*)d_out;

  {
    int wpb = 8; // waves per 256-thread block
    int blocks = (nsig + wpb - 1) / wpb;
    stats_kernel<<<blocks, 256, 0, stream>>>(x, meanv, invv, mxv, nsig, nt);
  }
  int waves = (nsig + 15) / 16;
  int wpb = 4; // waves per 128-thread block
  int blocks = (waves + wpb - 1) / wpb;
  fwd_kernel<<<blocks, 128, 0, stream>>>(x, b, a, taper, meanv, invv, y, nsig,
                                         nt);
  bwd_kernel<<<blocks, 128, 0, stream>>>(b, a, mxv, y, nsig, nt);
}